// MPSELoss_58935541235978
// MI455X (gfx1250) — compile-verified
//
#include <hip/hip_runtime.h>

// MPSE pairwise loss, algebraically reduced:
//   sum_{i<j} (d_i - d_j)^2  ==  N * sum(d^2) - (sum d)^2   per batch row.
// Row sums computed on the matrix core: S1 = d x ones, S2 = (d*d) x ones
// via V_WMMA_F32_16X16X4_F32 (full f32 precision).

typedef float v2f __attribute__((ext_vector_type(2)));
typedef float v8f __attribute__((ext_vector_type(8)));

#define N_COLS 2048
#define B_ROWS 32
#define WAVES 32              // 1024 threads, one block, one launch
#define WAVES_PER_SLAB 16     // 16 waves share one 16-row slab
#define COLS_PER_WAVE (N_COLS / WAVES_PER_SLAB)   // 128
#define CHUNKS (COLS_PER_WAVE / 4)                // 32 K-steps of 4

__global__ __launch_bounds__(1024)
void mpse_loss_wmma_kernel(const float* __restrict__ outputs,
                           const float* __restrict__ targets,
                           float* __restrict__ out) {
  __shared__ float lds1[WAVES][16];   // per-wave partial S1 per row
  __shared__ float lds2[WAVES][16];   // per-wave partial S2 per row
  __shared__ float rowloss[B_ROWS];

  const int tid  = threadIdx.x;
  const int lane = tid & 31;
  const int wave = tid >> 5;
  const int slab = wave >> 4;          // 0 -> rows 0..15, 1 -> rows 16..31
  const int cg   = wave & 15;          // column group within the slab
  const int m    = lane & 15;          // A-matrix row index (M = lane % 16)
  const int half = lane >> 4;          // lane-half selects K pair {0,1} vs {2,3}
  const int row  = slab * 16 + m;      // global batch row

  // Each lane loads a contiguous float2 = its (half,vgpr) K-slots of the chunk.
  const long base = (long)row * N_COLS + (long)cg * COLS_PER_WAVE + (half << 1);

  v8f acc1 = {};   // accumulates S1(m) replicated across 16 columns
  v8f acc2 = {};   // accumulates S2(m) replicated across 16 columns
  v2f ones; ones[0] = 1.0f; ones[1] = 1.0f;   // all-ones B matrix (layout-immune)

  for (int c = 0; c < CHUNKS; ++c) {
    const long idx = base + (long)c * 4;
    v2f o = *(const v2f*)(outputs + idx);
    v2f t = *(const v2f*)(targets + idx);
    v2f d;   d[0]   = o[0] - t[0];  d[1]   = o[1] - t[1];
    v2f dsq; dsq[0] = d[0] * d[0];  dsq[1] = d[1] * d[1];
    // D = A x ones + C : every column of D holds the running row-sum.
    acc1 = __builtin_amdgcn_wmma_f32_16x16x4_f32(
        false, d,   false, ones, (short)0, acc1, false, false);
    acc2 = __builtin_amdgcn_wmma_f32_16x16x4_f32(
        false, dsq, false, ones, (short)0, acc2, false, false);
  }

  // C/D layout: VGPR v, lane-half h -> row (v + 8h), column = lane%16.
  // All 16 columns are identical (B = ones), so lanes with column 0
  // (lanes 0 and 16) between them hold all 16 row results.
  if (m == 0) {
    const int rbase = half * 8;
#pragma unroll
    for (int v = 0; v < 8; ++v) {
      lds1[wave][rbase + v] = acc1[v];
      lds2[wave][rbase + v] = acc2[v];
    }
  }
  __syncthreads();

  // Fixed-order cross-wave reduction per batch row (deterministic).
  if (tid < B_ROWS) {
    const int s  = tid >> 4;
    const int mr = tid & 15;
    float s1 = 0.0f, s2 = 0.0f;
    for (int w = s * WAVES_PER_SLAB; w < (s + 1) * WAVES_PER_SLAB; ++w) {
      s1 += lds1[w][mr];
      s2 += lds2[w][mr];
    }
    rowloss[tid] = (float)N_COLS * s2 - s1 * s1;
  }
  __syncthreads();

  if (tid == 0) {
    float total = 0.0f;
    for (int b = 0; b < B_ROWS; ++b) total += rowloss[b];
    // count = 32 * 2048*2047/2 = 67,076,096 < 2^26 : exact in f32.
    const float count = (float)B_ROWS * (0.5f * (float)N_COLS * (float)(N_COLS - 1));
    out[0] = total / count;
  }
}

extern "C" void kernel_launch(void* const* d_in, const int* in_sizes, int n_in,
                              void* d_out, int out_size, void* d_ws, size_t ws_size,
                              hipStream_t stream) {
  (void)in_sizes; (void)n_in; (void)out_size; (void)d_ws; (void)ws_size;
  const float* outputs = (const float*)d_in[0];
  const float* targets = (const float*)d_in[1];
  float* out = (float*)d_out;
  mpse_loss_wmma_kernel<<<1, 1024, 0, stream>>>(outputs, targets, out);
}